// Transf_15290083574071
// MI455X (gfx1250) — compile-verified
//
#include <hip/hip_runtime.h>
#include <hip/hip_bf16.h>
#include <math.h>

#define EMBD  10
#define SEQ   2048
#define BATCH 4
#define VOCABN 32000
#define ROWS  (BATCH * SEQ)   // 8192 query rows

typedef __attribute__((ext_vector_type(2))) float v2f;
typedef __attribute__((ext_vector_type(8))) float v8f;

// ---------------------------------------------------------------------------
// Kernel 1: h = emb[x] + pos; Q = h@Wq.T, K = h@Wk.T, V = h@Wv.T
// One thread per (b,s) row. 300 FMAs/thread — trivial.
// ---------------------------------------------------------------------------
__global__ void __launch_bounds__(256)
embed_qkv_kernel(const int* __restrict__ x,
                 const float* __restrict__ emb,
                 const float* __restrict__ pos,
                 const float* __restrict__ Wq,
                 const float* __restrict__ Wk,
                 const float* __restrict__ Wv,
                 float* __restrict__ Q,
                 float* __restrict__ K,
                 float* __restrict__ V) {
    int tid = blockIdx.x * blockDim.x + threadIdx.x;
    if (tid >= ROWS) return;
    int s   = tid & (SEQ - 1);
    int tok = x[tid];

    float h[EMBD];
#pragma unroll
    for (int d = 0; d < EMBD; ++d)
        h[d] = emb[(size_t)tok * EMBD + d] + pos[(size_t)s * EMBD + d];

#pragma unroll
    for (int i = 0; i < EMBD; ++i) {
        float q = 0.f, k = 0.f, v = 0.f;
#pragma unroll
        for (int d = 0; d < EMBD; ++d) {
            q += h[d] * Wq[i * EMBD + d];
            k += h[d] * Wk[i * EMBD + d];
            v += h[d] * Wv[i * EMBD + d];
        }
        Q[(size_t)tid * EMBD + i] = q;
        K[(size_t)tid * EMBD + i] = k;
        V[(size_t)tid * EMBD + i] = v;
    }
}

// ---------------------------------------------------------------------------
// Kernel 2: causal flash attention, one wave32 per query row.
// Lanes stripe over t <= s with online softmax; shfl_xor tree merge at end.
// Masked (t>s) entries underflow to exactly 0 in the reference (exp(-3.16e8)),
// so skipping them is bit-equivalent for the softmax result.
// ---------------------------------------------------------------------------
__global__ void __launch_bounds__(256)
attn_kernel(const float* __restrict__ Q,
            const float* __restrict__ Kb,
            const float* __restrict__ Vb,
            float* __restrict__ O) {
    const int lane = threadIdx.x & 31;
    const int wave = threadIdx.x >> 5;
    const int row  = blockIdx.x * (blockDim.x >> 5) + wave;   // 0..8191
    const int s    = row & (SEQ - 1);
    const int base = row & ~(SEQ - 1);                        // b * SEQ

    float q[EMBD];
    const float* qp = Q + (size_t)row * EMBD;
#pragma unroll
    for (int d = 0; d < EMBD; ++d) q[d] = qp[d];

    float m = -3.0e38f, l = 0.f;
    float acc[EMBD];
#pragma unroll
    for (int d = 0; d < EMBD; ++d) acc[d] = 0.f;

    const float scale = 0.31622776601683794f;  // 1/sqrt(10)
    for (int t = lane; t <= s; t += 32) {
        const float* kp = Kb + (size_t)(base + t) * EMBD;
        float sc = 0.f;
#pragma unroll
        for (int d = 0; d < EMBD; ++d) sc += q[d] * kp[d];
        sc *= scale;

        float nm   = fmaxf(m, sc);
        float cold = __expf(m - nm);
        float p    = __expf(sc - nm);
        const float* vp = Vb + (size_t)(base + t) * EMBD;
        l = l * cold + p;
#pragma unroll
        for (int d = 0; d < EMBD; ++d) acc[d] = acc[d] * cold + p * vp[d];
        m = nm;
    }

    // merge 32 partial softmax states across the wave
#pragma unroll
    for (int off = 16; off >= 1; off >>= 1) {
        float om = __shfl_xor(m, off, 32);
        float ol = __shfl_xor(l, off, 32);
        float nm = fmaxf(m, om);
        float c1 = __expf(m - nm);
        float c2 = __expf(om - nm);
        l = l * c1 + ol * c2;
#pragma unroll
        for (int d = 0; d < EMBD; ++d) {
            float oa = __shfl_xor(acc[d], off, 32);
            acc[d] = acc[d] * c1 + oa * c2;
        }
        m = nm;
    }

    if (lane == 0) {
        float inv = 1.0f / l;
        float* op = O + (size_t)row * EMBD;
#pragma unroll
        for (int d = 0; d < EMBD; ++d) op[d] = acc[d] * inv;
    }
}

// ---------------------------------------------------------------------------
// Kernel 3: logits = O[8192x10] @ Wo.T[10x32000] via V_WMMA_F32_16X16X4_F32.
// K=10 padded to 12 -> 3 WMMAs per 16x16 tile. One tile per wave.
// A layout (16x4 f32): lane%16 = M, lane/16 selects K-pair {0,1}/{2,3}, v2f.
// B layout (4x16 f32): symmetric, lane%16 = N.
// D layout: VGPR r -> row r + 8*(lane/16), col = lane%16.
// Output stream (1.05 GB) uses non-temporal stores: bandwidth-bound path.
// k>=10 lanes get zeros via clamped addresses + selects (EXEC stays all-1s,
// required by WMMA).
// ---------------------------------------------------------------------------
__global__ void __launch_bounds__(256)
logits_wmma_kernel(const float* __restrict__ O,
                   const float* __restrict__ Wo,
                   float* __restrict__ out) {
    const int lane = threadIdx.x & 31;
    const int wave = threadIdx.x >> 5;
    const int nt = blockIdx.x * 8 + wave;     // 0..1999  (N tile)
    const int mt = blockIdx.y;                // 0..511   (M tile)
    const int lm = lane & 15;
    const int kg = (lane >> 4) << 1;          // 0 or 2

    const int mrow = mt * 16 + lm;
    const int ncol = nt * 16 + lm;

    v2f a[3], bmat[3];
#pragma unroll
    for (int c = 0; c < 3; ++c) {
        int k0 = c * 4 + kg;
        int k1 = k0 + 1;
        float a0 = O [(size_t)mrow * EMBD + (k0 < EMBD ? k0 : 0)];
        float a1 = O [(size_t)mrow * EMBD + (k1 < EMBD ? k1 : 0)];
        float b0 = Wo[(size_t)ncol * EMBD + (k0 < EMBD ? k0 : 0)];
        float b1 = Wo[(size_t)ncol * EMBD + (k1 < EMBD ? k1 : 0)];
        a[c].x    = (k0 < EMBD) ? a0 : 0.f;
        a[c].y    = (k1 < EMBD) ? a1 : 0.f;
        bmat[c].x = (k0 < EMBD) ? b0 : 0.f;
        bmat[c].y = (k1 < EMBD) ? b1 : 0.f;
    }

    v8f d = {};
#pragma unroll
    for (int c = 0; c < 3; ++c) {
        // 8 args: (neg_a, A, neg_b, B, c_mod, C, reuse_a, reuse_b)
        d = __builtin_amdgcn_wmma_f32_16x16x4_f32(
                false, a[c], false, bmat[c], (short)0, d, false, false);
    }

    const int    mbase   = mt * 16 + ((lane >> 4) << 3);
    const size_t colbase = (size_t)nt * 16 + lm;
#pragma unroll
    for (int r = 0; r < 8; ++r) {
        __builtin_nontemporal_store(
            d[r], out + (size_t)(mbase + r) * VOCABN + colbase);
    }
}

// ---------------------------------------------------------------------------
// Launch. Workspace layout (floats): Q | K | V | O  = 4 * 8192 * 10 floats
// = 1.31 MB total (well under any reasonable ws_size).
// ---------------------------------------------------------------------------
extern "C" void kernel_launch(void* const* d_in, const int* in_sizes, int n_in,
                              void* d_out, int out_size, void* d_ws, size_t ws_size,
                              hipStream_t stream) {
    (void)in_sizes; (void)n_in; (void)out_size; (void)ws_size;

    const int*   x   = (const int*)  d_in[0];
    const float* emb = (const float*)d_in[1];
    const float* pos = (const float*)d_in[2];
    const float* Wq  = (const float*)d_in[3];
    const float* Wk  = (const float*)d_in[4];
    const float* Wv  = (const float*)d_in[5];
    const float* Wo  = (const float*)d_in[6];
    float*       out = (float*)d_out;

    float* ws = (float*)d_ws;
    float* Q  = ws;
    float* K  = ws + (size_t)ROWS * EMBD;
    float* V  = ws + (size_t)2 * ROWS * EMBD;
    float* O  = ws + (size_t)3 * ROWS * EMBD;

    // 1) embeddings + QKV projections: 8192 rows, 1 thread/row
    embed_qkv_kernel<<<ROWS / 256, 256, 0, stream>>>(x, emb, pos, Wq, Wk, Wv,
                                                     Q, K, V);

    // 2) causal flash attention: 1 wave per row, 8 waves per block
    attn_kernel<<<ROWS / 8, 256, 0, stream>>>(Q, K, V, O);

    // 3) logits GEMM (WMMA): grid = (2000 N-tiles / 8 waves, 512 M-tiles)
    logits_wmma_kernel<<<dim3((VOCABN / 16) / 8, ROWS / 16), 256, 0, stream>>>(
        O, Wo, out);
}